// PulseShaping_63986422776108
// MI455X (gfx1250) — compile-verified
//
#include <hip/hip_runtime.h>

typedef __attribute__((ext_vector_type(2))) float v2f;
typedef __attribute__((ext_vector_type(8))) float v8f;

#define QBLOCK 128           // q positions per workgroup (8 waves * 16)
#define MAXS   16            // max K-steps -> filters up to 4*MAXS taps/phase
#define PADH   (16 * MAXS)   // per-region filter buffer (zero padded)
#define YBUFSZ (QBLOCK + 4 * MAXS)

// Polyphase pulse shaping via V_WMMA_F32_16X16X4_F32.
// out[4q+p] = sum_t h[4t+p] * y[q-t]  (complex y, complex h; h_imag==0 in practice)
__global__ __launch_bounds__(256)
void pulse_shape_wmma(const float* __restrict__ yr, const float* __restrict__ yi,
                      const float* __restrict__ hr, const float* __restrict__ hi,
                      float* __restrict__ out, int N, int L, int S, int outlen)
{
    __shared__ float hbuf[4 * PADH];                    // f=0: h_r, f=1: h_i, f=2,3: 0
    __shared__ float zbr[YBUFSZ];                       // reversed y_real tile
    __shared__ float zbi[YBUFSZ];                       // reversed y_imag tile
    __shared__ __align__(16) float obuf[2][4 * QBLOCK]; // staged outputs (coalesced store)

    const int tid = threadIdx.x;
    const int q0b = blockIdx.x * QBLOCK;

    // ---- Stage filter (zero-padded: idx >= L and regions f>=2 are 0) ----
    for (int i = tid; i < 4 * PADH; i += 256) {
        int f   = i / PADH;
        int idx = i - f * PADH;
        float v = 0.f;
        if (idx < L) {
            if (f == 0)      v = hr[idx];
            else if (f == 1) v = hi[idx];
        }
        hbuf[i] = v;
    }

    // ---- Stage y tile REVERSED: z[i] = y[Cbase - i], async HBM -> LDS ----
    const int halo  = 4 * S;
    const int ylen  = QBLOCK + halo;
    const int Cbase = q0b + QBLOCK - 1;
    for (int i = tid; i < ylen; i += 256) {
        int g = Cbase - i;
        if (g >= 0 && g < N) {
            unsigned l0 = (unsigned)(size_t)&zbr[i];
            unsigned l1 = (unsigned)(size_t)&zbi[i];
            unsigned long long a0 = (unsigned long long)(size_t)(yr + g);
            unsigned long long a1 = (unsigned long long)(size_t)(yi + g);
            asm volatile("global_load_async_to_lds_b32 %0, %1, off"
                         :: "v"(l0), "v"(a0) : "memory");
            asm volatile("global_load_async_to_lds_b32 %0, %1, off"
                         :: "v"(l1), "v"(a1) : "memory");
        } else {
            zbr[i] = 0.f;
            zbi[i] = 0.f;
        }
    }
    asm volatile("s_wait_asynccnt 0" ::: "memory");
    __syncthreads();

    // ---- WMMA polyphase chains ----
    const int lane = tid & 31;
    const int wave = tid >> 5;      // 0..7 -> one 16-q tile each
    const int half = lane >> 4;     // K/row split
    const int M    = lane & 15;     // A row
    const int kb   = half * 2;      // base k of this lane's two A/B VGPRs
    const int j    = lane & 15;     // B/C column
    const int p    = j & 3;         // phase
    const int f    = j >> 2;        // 0:h_r 1:h_i 2,3: zero region

    const int qoff  = wave * 16;
    // reversed layout: A[M,k] = y[q - (4s+k)] = z[abase + 4s + (k-kb)]
    const int abase = (QBLOCK - 1) - qoff - M + kb;
    const int bbase = f * PADH + 4 * kb + p;

    v8f cr = {};
    v8f ci = {};
    for (int s = 0; s < S; ++s) {
        const int ai = abase + 4 * s;
        const int bi = bbase + 16 * s;
        v2f a_r, a_i, b;
        a_r.x = zbr[ai];     a_r.y = zbr[ai + 1];   // ascending -> no swap movs
        a_i.x = zbi[ai];     a_i.y = zbi[ai + 1];
        b.x   = hbuf[bi];    b.y   = hbuf[bi + 4];
        cr = __builtin_amdgcn_wmma_f32_16x16x4_f32(false, a_r, false, b,
                                                   (short)0, cr, false, false);
        ci = __builtin_amdgcn_wmma_f32_16x16x4_f32(false, a_i, false, b,
                                                   (short)0, ci, false, false);
    }

    // ---- Combine complex parts (columns p and p+4 live in lanes j and j^4) ----
    #pragma unroll
    for (int v = 0; v < 8; ++v) {
        float crv  = cr[v];
        float civ  = ci[v];
        float cr_x = __shfl_xor(crv, 4, 32);  // C_r[i, j^4]
        float ci_x = __shfl_xor(civ, 4, 32);  // C_i[i, j^4]
        if (f == 0) {                          // lanes j<4 own phase p=j
            int i = v + 8 * half;              // row within tile
            obuf[0][qoff * 4 + 4 * i + p] = crv - ci_x;   // real
            obuf[1][qoff * 4 + 4 * i + p] = cr_x + civ;   // imag
        }
    }
    __syncthreads();

    // ---- Coalesced b128 stores: 256 threads x 4 floats = 1024 outputs ----
    const int c   = tid >> 7;            // 0: real plane, 1: imag plane
    const int idx = (tid & 127) * 4;
    const int m0  = 4 * q0b + idx;
    float* dst = out + (size_t)c * outlen;
    if (((outlen & 3) == 0) && (m0 + 3 < outlen)) {
        *(float4*)(dst + m0) = *(const float4*)&obuf[c][idx];
    } else {
        for (int e = 0; e < 4; ++e)
            if (m0 + e < outlen) dst[m0 + e] = obuf[c][idx + e];
    }
}

extern "C" void kernel_launch(void* const* d_in, const int* in_sizes, int n_in,
                              void* d_out, int out_size, void* d_ws, size_t ws_size,
                              hipStream_t stream) {
    const float* yr = (const float*)d_in[0];
    const float* yi = (const float*)d_in[1];
    const float* hr = (const float*)d_in[2];
    const float* hi = (const float*)d_in[3];
    // d_in[4] = n_up (==4); polyphase structure is specialized for 4.

    const int N = in_sizes[0];
    const int L = in_sizes[2];           // 129 taps
    float* out  = (float*)d_out;
    const int outlen = out_size / 2;     // 4N + L - 1 per component

    int T = (L + 3) / 4;                 // taps per phase (33)
    int S = (T + 3) / 4;                 // WMMA K=4 steps per chain (9)
    if (S > MAXS) S = MAXS;

    const int numQ   = ((outlen - 1) >> 2) + 1;
    const int blocks = (numQ + QBLOCK - 1) / QBLOCK;

    pulse_shape_wmma<<<blocks, 256, 0, stream>>>(yr, yi, hr, hi, out,
                                                 N, L, S, outlen);
}